// Update_entity_661424963868
// MI455X (gfx1250) — compile-verified
//
#include <hip/hip_runtime.h>
#include <cmath>

typedef __attribute__((ext_vector_type(2))) float v2f;
typedef __attribute__((ext_vector_type(4))) float v4f;
typedef __attribute__((ext_vector_type(8))) float v8f;

static constexpr int KC = 64;  // K-chunk staged into LDS per iteration
static constexpr int MT = 32;  // M rows per workgroup (2 sub-tiles per wave)

// ---------------------------------------------------------------------------
// Kernel 0: bulk copy hiddens -> out (so the indexed rows are an overwrite)
// ---------------------------------------------------------------------------
__global__ void copy_f4_kernel(const float4* __restrict__ src,
                               float4* __restrict__ dst, long long n4) {
  long long i = (long long)blockIdx.x * blockDim.x + threadIdx.x;
  if (i < n4) dst[i] = src[i];
}

// ---------------------------------------------------------------------------
// Kernel 1: Ws[b,e] = dot(s[b,:], W[e,:])   (s @ W^T), tiny GEMV-like pass
// ---------------------------------------------------------------------------
__global__ void ws_bias_kernel(const float* __restrict__ s,
                               const float* __restrict__ W,
                               float* __restrict__ wsb, int B, int D) {
  int i = blockIdx.x * blockDim.x + threadIdx.x;  // i = b*D + e
  if (i >= B * D) return;
  int b = i / D, e = i - b * D;
  const float* sr = s + (size_t)b * D;
  const float* wr = W + (size_t)e * D;
  float acc = 0.f;
  for (int d = 0; d < D; ++d) acc = fmaf(sr[d], wr[d], acc);
  wsb[i] = acc;
}

// ---------------------------------------------------------------------------
// Kernel 2: g[b,k] = sigmoid(dot(s[b,:], h[b,k,:] + w[b,k,:]))
// ---------------------------------------------------------------------------
__global__ void gate_kernel(const float* __restrict__ s,
                            const float* __restrict__ hiddens,
                            const float* __restrict__ keys,
                            const int* __restrict__ idx,
                            float* __restrict__ g, int B, int N, int D, int K) {
  int R = blockIdx.x * blockDim.x + threadIdx.x;  // R = b*K + k
  if (R >= B * K) return;
  int b = R / K, kk = R - b * K;
  int ni = idx[kk];
  const float* sr = s + (size_t)b * D;
  const float* hr = hiddens + ((size_t)b * N + ni) * D;
  const float* wr = keys + ((size_t)b * N + ni) * D;
  float acc = 0.f;
  for (int d = 0; d < D; ++d) acc = fmaf(sr[d], hr[d] + wr[d], acc);
  g[R] = 1.f / (1.f + __expf(-acc));
}

// ---------------------------------------------------------------------------
// Kernel 3: fused  T = relu(h@U^T + w@V^T + Ws) ; hn = h + g*T ; L2-normalize
// One workgroup = 32 output rows x full D columns.  16 wave32s; each wave
// computes a 32x64 slab as 2x4 accumulators of v8f via v_wmma_f32_16x16x4_f32,
// chaining both GEMMs into the same accumulators.  Each B fragment (U/V)
// feeds 2 WMMAs (M reuse), halving L2 traffic vs a 16-row tile.
// ---------------------------------------------------------------------------
__global__ void __launch_bounds__(512)
update_wmma_kernel(const float* __restrict__ hiddens,
                   const float* __restrict__ keys,
                   const float* __restrict__ U,
                   const float* __restrict__ V,
                   const int* __restrict__ idx,
                   const float* __restrict__ wsb,   // [B,D]
                   const float* __restrict__ gate,  // [B*K]
                   float* __restrict__ out,         // [B,N,D]
                   int B, int N, int D, int K) {
  __shared__ float lds_h[MT * KC];
  __shared__ float lds_w[MT * KC];
  __shared__ float lds_ss[MT];

  const int tid     = threadIdx.x;
  const int lane    = tid & 31;
  const int wv      = tid >> 5;      // wave id 0..15
  const int laneLow = lane & 15;     // N (or M) index inside fragment
  const int laneHi  = lane >> 4;     // selects K pair {0,1} vs {2,3}
  const int rowsTot = B * K;

  const int R0 = blockIdx.x * MT;    // first global row of this M-tile

  // Cooperative-staging assignment: thread -> (row, 4 columns) of the A tiles
  const int srow = tid >> 4;             // 0..31
  const int scol = (tid & 15) * 4;       // 0..60
  int Rs = R0 + srow; if (Rs >= rowsTot) Rs = rowsTot - 1;
  const int bS = Rs / K, kS = Rs - bS * K;
  const int niS = idx[kS];
  const float* hrow = hiddens + ((size_t)bS * N + niS) * D;
  const float* wrow = keys    + ((size_t)bS * N + niS) * D;

  const int n0w = wv * 64;               // this wave's first output column

  v8f acc[2][4];
  const v8f vzero = {0.f, 0.f, 0.f, 0.f, 0.f, 0.f, 0.f, 0.f};
#pragma unroll
  for (int p = 0; p < 2; ++p)
#pragma unroll
    for (int s2 = 0; s2 < 4; ++s2) acc[p][s2] = vzero;

  for (int k0 = 0; k0 < D; k0 += KC) {
    __syncthreads();
    // Stage MTxKC tiles of h and w into LDS (shared A operands for all waves)
    *(v4f*)&lds_h[srow * KC + scol] = *(const v4f*)&hrow[k0 + scol];
    *(v4f*)&lds_w[srow * KC + scol] = *(const v4f*)&wrow[k0 + scol];
    __syncthreads();

#pragma unroll 4
    for (int kk = 0; kk < KC; kk += 4) {
      const int ko = kk + 2 * laneHi;
      // A fragments (16x4 f32): lane = M row; {x,y} = K pair chosen by laneHi
      v2f ah[2], aw[2];
      ah[0] = *(const v2f*)&lds_h[laneLow * KC + ko];
      ah[1] = *(const v2f*)&lds_h[(16 + laneLow) * KC + ko];
      aw[0] = *(const v2f*)&lds_w[laneLow * KC + ko];
      aw[1] = *(const v2f*)&lds_w[(16 + laneLow) * KC + ko];
#pragma unroll
      for (int s2 = 0; s2 < 4; ++s2) {
        const int e = n0w + s2 * 16 + laneLow;  // output column = U/V row
        // B fragment (4x16 f32): B[k][n] = U[e=n][d=k] -> contiguous float2
        v2f bu = *(const v2f*)&U[(size_t)e * D + (k0 + ko)];
        v2f bv = *(const v2f*)&V[(size_t)e * D + (k0 + ko)];
#pragma unroll
        for (int p = 0; p < 2; ++p) {
          acc[p][s2] = __builtin_amdgcn_wmma_f32_16x16x4_f32(
              false, ah[p], false, bu, (short)0, acc[p][s2], false, false);
          acc[p][s2] = __builtin_amdgcn_wmma_f32_16x16x4_f32(
              false, aw[p], false, bv, (short)0, acc[p][s2], false, false);
        }
      }
    }
  }

  __syncthreads();
  if (tid < MT) lds_ss[tid] = 0.f;
  __syncthreads();

  // Epilogue pass 1: hn = h + g * relu(acc + Ws) ; accumulate row sum-squares
#pragma unroll
  for (int p = 0; p < 2; ++p) {
#pragma unroll
    for (int s2 = 0; s2 < 4; ++s2) {
#pragma unroll
      for (int r = 0; r < 8; ++r) {
        const int m = p * 16 + laneHi * 8 + r;   // local row 0..31
        int R = R0 + m; if (R >= rowsTot) R = rowsTot - 1;
        const int b = R / K, kr = R - b * K;
        const int ni = idx[kr];
        const int e = n0w + s2 * 16 + laneLow;
        float t = acc[p][s2][r] + wsb[(size_t)b * D + e];
        t = fmaxf(t, 0.f);
        const float h0 = hiddens[((size_t)b * N + ni) * D + e];
        const float hn = fmaf(gate[R], t, h0);
        acc[p][s2][r] = hn;
        atomicAdd(&lds_ss[m], hn * hn);          // ds_add_f32
      }
    }
  }
  __syncthreads();

  // Epilogue pass 2: normalize and scatter into out[:, indices, :]
#pragma unroll
  for (int p = 0; p < 2; ++p) {
#pragma unroll
    for (int s2 = 0; s2 < 4; ++s2) {
#pragma unroll
      for (int r = 0; r < 8; ++r) {
        const int m = p * 16 + laneHi * 8 + r;
        int R = R0 + m; if (R >= rowsTot) R = rowsTot - 1;
        const int b = R / K, kr = R - b * K;
        const int ni = idx[kr];
        const int e = n0w + s2 * 16 + laneLow;
        const float scale = rsqrtf(fmaxf(lds_ss[m], 1e-12f));
        out[((size_t)b * N + ni) * D + e] = acc[p][s2][r] * scale;
      }
    }
  }
}

// ---------------------------------------------------------------------------
extern "C" void kernel_launch(void* const* d_in, const int* in_sizes, int n_in,
                              void* d_out, int out_size, void* d_ws, size_t ws_size,
                              hipStream_t stream) {
  const float* s       = (const float*)d_in[0];
  const float* hiddens = (const float*)d_in[1];
  const float* keys    = (const float*)d_in[2];
  const float* U       = (const float*)d_in[3];
  const float* V       = (const float*)d_in[4];
  const float* W       = (const float*)d_in[5];
  const int*   idx     = (const int*)d_in[6];
  float*       out     = (float*)d_out;

  // Derive shapes: U is DxD, s is BxD, hiddens is BxNxD, indices is K
  const int D = (int)llround(sqrt((double)in_sizes[3]));
  const int B = in_sizes[0] / D;
  const int N = (int)((long long)in_sizes[1] / ((long long)B * D));
  const int K = in_sizes[6];

  float* wsb = (float*)d_ws;                    // [B*D]
  float* g   = wsb + (size_t)B * D;             // [B*K]

  // 0) hiddens -> out (float4)
  {
    long long n4 = ((long long)B * N * D) / 4;
    int grid = (int)((n4 + 255) / 256);
    copy_f4_kernel<<<grid, 256, 0, stream>>>((const float4*)hiddens,
                                             (float4*)out, n4);
  }
  // 1) Ws = s @ W^T
  {
    int n = B * D;
    ws_bias_kernel<<<(n + 255) / 256, 256, 0, stream>>>(s, W, wsb, B, D);
  }
  // 2) g = sigmoid(<s, h + w>)
  {
    int n = B * K;
    gate_kernel<<<(n + 255) / 256, 256, 0, stream>>>(s, hiddens, keys, idx, g,
                                                     B, N, D, K);
  }
  // 3) fused WMMA update + normalize + scatter
  {
    int grid = (B * K + MT - 1) / MT;
    update_wmma_kernel<<<grid, 512, 0, stream>>>(hiddens, keys, U, V, idx, wsb,
                                                 g, out, B, N, D, K);
  }
}